// tree_crf_15685220565496
// MI455X (gfx1250) — compile-verified
//
#include <hip/hip_runtime.h>
#include <hip/hip_bf16.h>
#include <stdint.h>

#ifndef __has_builtin
#define __has_builtin(x) 0
#endif

// Problem constants (from reference setup_inputs): B=40, M=128, R=32.
#define TB 40
#define TM 128
#define TR 32
#define TMM (TM * TM)          // 16384 (i,j) pairs per batch
#define NTH 512                // 16 wave32 waves per workgroup
#define NEG_INF_F (-__builtin_inff())

// LDS layout for the decode kernel (dynamic shared memory)
#define OFF_E2     0                     // 16384 floats  = 65536 B
#define OFF_PART   (TMM * 4)             // 128 ints      =   512 B
#define OFF_WK     (OFF_PART + 512)      // 17 u64 keys (pad 256 B)
#define OFF_SCORE  (OFF_WK + 256)        // 128 floats    =   512 B
#define SMEM_BYTES (OFF_SCORE + 512)     // 66816 B  (<< 320 KB/WGP on CDNA5)

// Staging-path selection (device pass only; host pass falls to plain copy)
#if __has_builtin(__builtin_amdgcn_tensor_load_to_lds) && __has_builtin(__builtin_amdgcn_s_wait_tensorcnt)
#define USE_TDM 1
typedef __attribute__((ext_vector_type(4))) unsigned int u32x4;
typedef __attribute__((ext_vector_type(8))) int i32x8;
typedef __attribute__((ext_vector_type(4))) int i32x4;
#elif __has_builtin(__builtin_amdgcn_global_load_async_to_lds_b128)
#define USE_ASYNC 1
#endif

// Order-preserving float<->uint transform: f2ord is strictly monotone, so
// u64 key = (f2ord(val)<<32) | ~idx makes "max key" == "max val, then min idx"
// (exactly jnp.argmax first-max tie-breaking). Round-trips bits exactly.
__device__ __forceinline__ unsigned f2ord(float f) {
    unsigned u = __float_as_uint(f);
    return (u & 0x80000000u) ? ~u : (u | 0x80000000u);
}
__device__ __forceinline__ float ord2f(unsigned o) {
    unsigned u = (o & 0x80000000u) ? (o & 0x7fffffffu) : ~o;
    return __uint_as_float(u);
}

// ---------------------------------------------------------------------------
// Kernel 1: E2[b,i,j] = max_r logE[b,i,j,r].  Streams the full 84 MB once
// (the bandwidth floor of this problem: ~3.6 us at 23.3 TB/s). Each thread
// owns one contiguous 128-byte run of R=32 floats -> 8x global_load_b128.
// ---------------------------------------------------------------------------
__global__ __launch_bounds__(256)
void k_rmax(const float* __restrict__ e, float* __restrict__ e2)
{
    int idx = blockIdx.x * 256 + threadIdx.x;
    if (idx >= TB * TMM) return;
    const float4* p = (const float4*)(e + (size_t)idx * TR);
    // gfx1250 software prefetch a few KB ahead of the stream
    __builtin_prefetch((const void*)(p + 256), 0, 1);
    float m = NEG_INF_F;
#pragma unroll
    for (int k = 0; k < TR / 4; ++k) {
        float4 v = p[k];
        m = fmaxf(m, fmaxf(fmaxf(v.x, v.y), fmaxf(v.z, v.w)));
    }
    e2[idx] = m;
}

// ---------------------------------------------------------------------------
// Kernel 2: greedy partition-suppression MST decode, one workgroup per batch.
// E2 tile (64 KB) staged into LDS via TDM / async-LDS; 128 serial steps of a
// masked argmax over 16K LDS floats. Validity needs NO per-element mask:
//   edge (i,j), i!=j : valid iff part[i] != part[j]
//   edge (i,i)       : valid unless diagonal was killed (self-merge) --
//                      handled by writing -inf directly into E2 LDS
//   row lf           : killed by writing 128x -inf after each pick
// Also computes the label score and writes partial[b] = mst_acc - score.
// ---------------------------------------------------------------------------
__global__ __launch_bounds__(NTH)
void k_mst(const float* __restrict__ e2g,
           const float* __restrict__ logE,
           const int*   __restrict__ tree,
           float*       __restrict__ partial)
{
    extern __shared__ char smem[];
    float* E2   = (float*)(smem + OFF_E2);
    int*   part = (int*)  (smem + OFF_PART);
    unsigned long long* wk = (unsigned long long*)(smem + OFF_WK);
    float* sscore = (float*)(smem + OFF_SCORE);

    const int b = blockIdx.x;
    const int t = threadIdx.x;
    const float* __restrict__ src = e2g + (size_t)b * TMM;

    // ---- stage E2 -> LDS ---------------------------------------------------
#if defined(USE_TDM)
    if (t < 32) {  // wave 0 only: one TDM moves the whole 64 KB tile
        unsigned long long ga = (unsigned long long)(uintptr_t)src;
        unsigned lds_off = (unsigned)(uintptr_t)E2;  // low 32 bits = LDS offset
        // D# group0: count=1 valid, lds_addr, global_addr, type=2 ("image")
        u32x4 g0 = { 1u, lds_off, (unsigned)ga,
                     (unsigned)(ga >> 32) | 0x80000000u };
        // D# group1: data_size=4B; 128x128 tensor, 128x128 tile, stride0=128
        i32x8 g1 = { 0x20000, TM << 16, TM << 16, TM << 16, TM, TM, 0, 0 };
        i32x4 gz4 = { 0, 0, 0, 0 };
        i32x8 gz8 = { 0, 0, 0, 0, 0, 0, 0, 0 };
        __builtin_amdgcn_tensor_load_to_lds(g0, g1, gz4, gz4, gz8, 0);
        __builtin_amdgcn_s_wait_tensorcnt(0);
    }
#elif defined(USE_ASYNC)
#pragma unroll
    for (int k = 0; k < (TMM / 4) / NTH; ++k) {
        int c = k * NTH + t;
        __builtin_amdgcn_global_load_async_to_lds_b128(
            (__attribute__((address_space(1))) void*)(src + c * 4),
            (__attribute__((address_space(3))) void*)(E2 + c * 4),
            0, 0);
    }
#if __has_builtin(__builtin_amdgcn_s_wait_asynccnt)
    __builtin_amdgcn_s_wait_asynccnt(0);
#else
    asm volatile("s_wait_asynccnt 0x0" ::: "memory");
#endif
#else
    for (int c = t; c < TMM / 4; c += NTH)
        ((float4*)E2)[c] = ((const float4*)src)[c];
#endif

    // ---- init partition labels + label-score gather (overlaps the DMA) -----
    if (t < TM) {
        part[t] = t;
        const int* tp = tree + ((size_t)b * TM + t) * 3;
        int lf = tp[0], lt = tp[1], r = tp[2];
        bool valid = !((lf == -1) && (lt == -1) && (r == -1));
        int clf = lf > 0 ? lf : 0;
        int clt = lt > 0 ? lt : 0;
        int cr  = r  > 0 ? r  : 0;
        float v = logE[(((size_t)b * TM + clf) * TM + clt) * TR + cr];
        sscore[t] = valid ? v : 0.0f;
    }
    __syncthreads();

    // ---- 128 greedy steps --------------------------------------------------
    float acc = 0.0f;
    for (int step = 0; step < TM; ++step) {
        // masked scan: 8 float4 per thread, validity from partition labels
        float bestv = NEG_INF_F;
        int   besti = 0x7FFFFFFF;
#pragma unroll
        for (int k = 0; k < (TMM / 4) / NTH; ++k) {
            int c = k * NTH + t;           // float4 index; stride-1 across lanes
            float4 v = ((const float4*)E2)[c];
            int idx = c * 4;
            int i = idx >> 7;
            int j = idx & 127;             // multiple of 4
            int pi = part[i];
            int4 pj = ((const int4*)part)[c & 31];
            bool v0 = (pi != pj.x) || (i == j);
            bool v1 = (pi != pj.y) || (i == j + 1);
            bool v2 = (pi != pj.z) || (i == j + 2);
            bool v3 = (pi != pj.w) || (i == j + 3);
            if (v0 && (v.x > bestv || (v.x == bestv && idx     < besti))) { bestv = v.x; besti = idx;     }
            if (v1 && (v.y > bestv || (v.y == bestv && idx + 1 < besti))) { bestv = v.y; besti = idx + 1; }
            if (v2 && (v.z > bestv || (v.z == bestv && idx + 2 < besti))) { bestv = v.z; besti = idx + 2; }
            if (v3 && (v.w > bestv || (v.w == bestv && idx + 3 < besti))) { bestv = v.w; besti = idx + 3; }
        }
        // pack (val,idx) into a monotone u64 key; wave32 xor-shuffle reduce
        unsigned long long key =
            ((unsigned long long)f2ord(bestv) << 32) | (unsigned)(~(unsigned)besti);
#pragma unroll
        for (int m = 16; m > 0; m >>= 1) {
            unsigned lo = (unsigned)key, hi = (unsigned)(key >> 32);
            unsigned olo = __shfl_xor(lo, m, 32);
            unsigned ohi = __shfl_xor(hi, m, 32);
            unsigned long long o = ((unsigned long long)ohi << 32) | olo;
            if (o > key) key = o;
        }
        if ((t & 31) == 0) wk[t >> 5] = key;
        __syncthreads();
        if (t == 0) {
            unsigned long long w = wk[0];
            for (int q = 1; q < NTH / 32; ++q) if (wk[q] > w) w = wk[q];
            wk[16] = w;
        }
        __syncthreads();
        unsigned long long w = wk[16];
        float wval = ord2f((unsigned)(w >> 32));
        if (!(wval > NEG_INF_F)) break;    // all remaining edges suppressed
        int widx = (int)~(unsigned)(w & 0xffffffffu);
        acc += wval;                        // exact reference accumulation order
        int lf = widx >> 7, lt = widx & 127;
        int pf = part[lf], pt = part[lt];   // read labels before anyone writes
        __syncthreads();
        if (t < TM) {
            int p = part[t];
            if (p == pf || p == pt) part[t] = pt;          // merge partitions
            if (lf == lt && p == pf) E2[t * TM + t] = NEG_INF_F;  // self-merge: kill diagonals of P
            E2[lf * TM + t] = NEG_INF_F;                   // kill row lf
        }
        __syncthreads();
    }

    if (t == 0) {
        float s = 0.0f;
        for (int q = 0; q < TM; ++q) s += sscore[q];
        partial[b] = acc - s;
    }
}

// ---------------------------------------------------------------------------
// Kernel 3: deterministic fixed-order mean over the 40 batch partials.
// ---------------------------------------------------------------------------
__global__ void k_mean(const float* __restrict__ partial, float* __restrict__ out)
{
    if (threadIdx.x == 0) {
        float s = 0.0f;
        for (int q = 0; q < TB; ++q) s += partial[q];
        out[0] = s / (float)TB;
    }
}

extern "C" void kernel_launch(void* const* d_in, const int* in_sizes, int n_in,
                              void* d_out, int out_size, void* d_ws, size_t ws_size,
                              hipStream_t stream) {
    const float* logE = (const float*)d_in[0];   // [40,128,128,32] fp32
    const int*   tree = (const int*)d_in[1];     // [40,128,3] int32
    float* out = (float*)d_out;                  // scalar fp32

    // workspace: E2 (40*16384 floats = 2.62 MB) + per-batch partials (40 floats)
    float* e2 = (float*)d_ws;
    float* partial = e2 + (size_t)TB * TMM;

    int total = TB * TMM;
    k_rmax<<<(total + 255) / 256, 256, 0, stream>>>(logE, e2);
    k_mst<<<TB, NTH, SMEM_BYTES, stream>>>(e2, logE, tree, partial);
    k_mean<<<1, 32, 0, stream>>>(partial, out);
}